// DepthwiseIIR_76020921139415
// MI455X (gfx1250) — compile-verified
//
#include <hip/hip_runtime.h>

// DepthwiseIIR (row recurrence over H) + BatchNorm(eval) + clamp(-8,8)
// x: [B=4, C=64, H=512, W=512] fp32.
//
// Recurrence blocked into 16-row tiles, each evaluated as a 16x16
// lower-triangular matmul on the CDNA5 matrix pipe:
//   y[h0+j] = (M @ Xc)[j] + wo^j * (fc*F + I),   M = wc*Id + fc*Lstrict
// via 4 chained V_WMMA_F32_16X16X4_F32 (K slices 0..3,4..7,8..11,12..15).
// Carries:  F' = wo^16*F + sum_m wo^{15-m} Xc[m]  (VALU dot + shfl_xor 16)
//           I' = wo^16*I
// One wave = one (b,c) plane x one 16-wide w strip, 32 tiles down H.
// Software pipeline depth 2 (3 live tile buffers) so each wave keeps
// 2 KB of loads in flight: 8192 waves x 2KB = 16 MB outstanding,
// above the ~14 MB latency-bandwidth product of 23.3 TB/s HBM.

#define BB 4
#define CC 64
#define HH 512
#define WW 512
#define EPSF 1e-3f

typedef __attribute__((ext_vector_type(2))) float v2f;
typedef __attribute__((ext_vector_type(8))) float v8f;

// wo^e for runtime e in [0,15] via binary decomposition (4 sel + 3 mul).
__device__ __forceinline__ float wo_pow(float wo, float wo2, float wo4,
                                        float wo8, int e) {
    float p = (e & 1) ? wo : 1.0f;
    p *= (e & 2) ? wo2 : 1.0f;
    p *= (e & 4) ? wo4 : 1.0f;
    p *= (e & 8) ? wo8 : 1.0f;
    return p;
}

__global__ __launch_bounds__(256) void DepthwiseIIR_wmma_kernel(
    const float* __restrict__ x,
    const float* __restrict__ w_curr,
    const float* __restrict__ w_prev_inp,
    const float* __restrict__ w_prev_out,
    const float* __restrict__ gamma,
    const float* __restrict__ beta,
    const float* __restrict__ rmean,
    const float* __restrict__ rvar,
    float* __restrict__ out)
{
    const int lane   = threadIdx.x & 31;
    const int lane15 = lane & 15;
    const int hi     = lane >> 4;                  // half-wave select
    const int gwave  = (blockIdx.x * blockDim.x + threadIdx.x) >> 5;

    const int strip = gwave & 31;                  // 32 strips of 16 w
    const int plane = gwave >> 5;                  // b*C + c, 0..255
    const int c     = plane & (CC - 1);
    const int w0    = strip * 16;
    const int colb  = w0 + lane15;                 // this lane's w column

    // ---- per-channel coefficients -------------------------------------
    const float wc  = w_curr[c];
    const float wi  = w_prev_inp[c];
    const float wo  = w_prev_out[c];
    const float fc  = wi + wo * wc;                           // feedback coeff
    const float inv = gamma[c] * rsqrtf(rvar[c] + EPSF);      // BN scale
    const float bns = beta[c] - rmean[c] * inv;               // BN shift

    const float wo2  = wo * wo;
    const float wo4  = wo2 * wo2;
    const float wo8  = wo4 * wo4;
    const float wo16 = wo8 * wo8;

    const float* xp = x   + (size_t)plane * HH * WW;
    float*       yp = out + (size_t)plane * HH * WW;

    // ---- A matrix (M = wc*Id + fc*Lstrict), 4 K-slices of 16x4 --------
    // A layout (32-bit, 16x4): row M = lane15, K = 2*hi + j  (j = vgpr)
    v2f A[4];
    v2f CD[4];   // carry-dot coefficients wo^(15-m) matching B layout rows
#pragma unroll
    for (int s = 0; s < 4; ++s) {
#pragma unroll
        for (int j = 0; j < 2; ++j) {
            const int m   = 4 * s + 2 * hi + j;   // column (K index)
            const int row = lane15;
            float val = 0.0f;
            if (m == row)      val = wc;
            else if (m < row)  val = fc * wo_pow(wo, wo2, wo4, wo8, row - 1 - m);
            A[s][j]  = val;
            CD[s][j] = wo_pow(wo, wo2, wo4, wo8, 15 - m);
        }
    }

    // per-D-VGPR row power wo^(v + 8*hi)
    float rp[8];
#pragma unroll
    for (int v = 0; v < 8; ++v) rp[v] = wo_pow(wo, wo2, wo4, wo8, v + 8 * hi);

    // ---- carries: F = f_{h0} (starts 0), I = ict_0 = (wi+wo)*x[0][w] ---
    float F = 0.0f;
    float I = (wi + wo) * xp[colb];

    // ---- tile loads in B layout ---------------------------------------
    // B layout (32-bit, 4x16 per slice): K = 2*hi + j, N = lane15
    // B element (slice s, vgpr j) = x[h0 + 4s + 2hi + j][w0 + lane15]
    const int krowbase = 2 * hi;

    v2f B0[4], B1[4], B2[4];
#pragma unroll
    for (int s = 0; s < 4; ++s) {
        B0[s][0] = xp[(size_t)(      4 * s + krowbase + 0) * WW + colb];
        B0[s][1] = xp[(size_t)(      4 * s + krowbase + 1) * WW + colb];
    }
#pragma unroll
    for (int s = 0; s < 4; ++s) {
        B1[s][0] = xp[(size_t)(16 +  4 * s + krowbase + 0) * WW + colb];
        B1[s][1] = xp[(size_t)(16 +  4 * s + krowbase + 1) * WW + colb];
    }

    for (int k = 0; k < 32; ++k) {
        const int h0 = k * 16;

        // prefetch distance 2: issue tile k+2 loads before tile k's math
        if (k < 30) {
#pragma unroll
            for (int s = 0; s < 4; ++s) {
                B2[s][0] = xp[(size_t)(h0 + 32 + 4 * s + krowbase + 0) * WW + colb];
                B2[s][1] = xp[(size_t)(h0 + 32 + 4 * s + krowbase + 1) * WW + colb];
            }
        }

        // D = M @ Xc  via 4 chained f32 WMMAs (K-slice accumulation)
        v8f D = {};
#pragma unroll
        for (int s = 0; s < 4; ++s) {
            D = __builtin_amdgcn_wmma_f32_16x16x4_f32(
                    /*neg_a=*/false, A[s], /*neg_b=*/false, B0[s],
                    /*c_mod=*/(short)0, D, /*reuse_a=*/false, /*reuse_b=*/false);
        }

        // carry dot: sum_m wo^(15-m) * Xc[m][w]  (half-wave partials + xor-16)
        float part = 0.0f;
#pragma unroll
        for (int s = 0; s < 4; ++s) {
            part = fmaf(CD[s][0], B0[s][0], part);
            part = fmaf(CD[s][1], B0[s][1], part);
        }
        part += __shfl_xor(part, 16, 32);

        // y[h0+j] = D[j] + wo^j * G ; then BN + clamp; store in C/D layout
        const float G = fmaf(fc, F, I);
#pragma unroll
        for (int v = 0; v < 8; ++v) {
            float y = fmaf(rp[v], G, D[v]);
            y = fmaf(y, inv, bns);
            y = fminf(8.0f, fmaxf(-8.0f, y));
            yp[(size_t)(h0 + v + 8 * hi) * WW + colb] = y;
        }

        // advance carries and rotate tile buffers
        F = fmaf(wo16, F, part);
        I *= wo16;
#pragma unroll
        for (int s = 0; s < 4; ++s) { B0[s] = B1[s]; B1[s] = B2[s]; }
    }
}

extern "C" void kernel_launch(void* const* d_in, const int* in_sizes, int n_in,
                              void* d_out, int out_size, void* d_ws, size_t ws_size,
                              hipStream_t stream) {
    const float* x          = (const float*)d_in[0];
    const float* w_curr     = (const float*)d_in[1];
    const float* w_prev_inp = (const float*)d_in[2];
    const float* w_prev_out = (const float*)d_in[3];
    const float* gamma      = (const float*)d_in[4];
    const float* beta       = (const float*)d_in[5];
    const float* rmean      = (const float*)d_in[6];
    const float* rvar       = (const float*)d_in[7];
    float* out              = (float*)d_out;

    const int waves   = (BB * CC) * (WW / 16);     // 8192 waves
    const int threads = 256;                        // 8 waves / block
    const int blocks  = waves * 32 / threads;       // 1024 blocks

    DepthwiseIIR_wmma_kernel<<<dim3(blocks), dim3(threads), 0, stream>>>(
        x, w_curr, w_prev_inp, w_prev_out, gamma, beta, rmean, rvar, out);
}